// LS_loss_67422396613173
// MI455X (gfx1250) — compile-verified
//
#include <hip/hip_runtime.h>

#define HWV 262144                 // 512*512 elements per (b,c) channel
#define NCH 128                    // 32*4 channels
#define BLOCKS_PER_CH 32
#define THREADS 256
#define V4_PER_TH (HWV / 4 / (BLOCKS_PER_CH * THREADS))   // 8 float4 per thread
#define EPS_F 1e-4f
#define SIG_K 57.70780163555852f   // 40 * log2(e):  0.5*(1+tanh(x/0.05)) == 1/(1+2^(-SIG_K*x))

typedef __attribute__((ext_vector_type(2))) float v2f;
typedef __attribute__((ext_vector_type(8))) float v8f;

__device__ __forceinline__ float heaviside(float x) {
  // 0.5*(1+tanh(x/0.05)) == sigmoid(40x); one v_exp_f32 + one v_rcp_f32
  float e = __builtin_amdgcn_exp2f(-SIG_K * x);
  return __builtin_amdgcn_rcpf(1.0f + e);
}

// ---- kernel 0: zero the 128x4 accumulator table (ws is poisoned, not re-zeroed) ----
__global__ void zero_ws_kernel(float* __restrict__ ws) {
  ws[threadIdx.x] = 0.0f;          // launched with NCH*4 = 512 threads
}

// ---- kernel 1: bandwidth-bound streaming pass; 4 partial sums per channel ----
__global__ void __launch_bounds__(THREADS)
partial_sums_kernel(const float* __restrict__ inp, const float* __restrict__ img,
                    float* __restrict__ ws) {
  const int ch  = blockIdx.x >> 5;         // / BLOCKS_PER_CH
  const int blk = blockIdx.x & 31;         // % BLOCKS_PER_CH
  const float4* in4 = (const float4*)inp + (size_t)ch * (HWV / 4);
  const float4* im4 = (const float4*)img + (size_t)ch * (HWV / 4);
  const int base = blk * (THREADS * V4_PER_TH) + threadIdx.x;

  float sH = 0.0f, sHI = 0.0f, sI = 0.0f, sI2 = 0.0f;
#pragma unroll
  for (int i = 0; i < V4_PER_TH; ++i) {
    const float4 x = in4[base + i * THREADS];   // global_load_b128
    const float4 m = im4[base + i * THREADS];   // global_load_b128
    const float h0 = heaviside(x.x), h1 = heaviside(x.y);
    const float h2 = heaviside(x.z), h3 = heaviside(x.w);
    sH  += (h0 + h1) + (h2 + h3);
    sHI += h0 * m.x + h1 * m.y + h2 * m.z + h3 * m.w;
    sI  += (m.x + m.y) + (m.z + m.w);
    sI2 += m.x * m.x + m.y * m.y + m.z * m.z + m.w * m.w;
  }

  // wave32 tree reduction
  for (int off = 16; off > 0; off >>= 1) {
    sH  += __shfl_down(sH,  off);
    sHI += __shfl_down(sHI, off);
    sI  += __shfl_down(sI,  off);
    sI2 += __shfl_down(sI2, off);
  }

  __shared__ float lds[THREADS / 32][4];
  const int wave = threadIdx.x >> 5, lane = threadIdx.x & 31;
  if (lane == 0) {
    lds[wave][0] = sH; lds[wave][1] = sHI; lds[wave][2] = sI; lds[wave][3] = sI2;
  }
  __syncthreads();
  if (threadIdx.x < 4) {
    float s = 0.0f;
#pragma unroll
    for (int w = 0; w < THREADS / 32; ++w) s += lds[w][threadIdx.x];
    atomicAdd(&ws[ch * 4 + threadIdx.x], s);   // global_atomic_add_f32
  }
}

// ---- kernel 2: per-channel closed-form loss + WMMA f32 grand-total reduction ----
__global__ void finalize_kernel(const float* __restrict__ ws, float* __restrict__ out) {
  const int lane = threadIdx.x;              // exactly one wave32, EXEC all ones
  const float HWf = (float)HWV;

  // 128 channel losses folded to 64 values; each lane supplies 2 A-matrix slots.
  float fold[2];
#pragma unroll
  for (int j = 0; j < 2; ++j) {
    const int idx = ((lane & 15) << 2) + ((lane >> 4) << 1) + j;  // 0..63, bijective
    float f = 0.0f;
#pragma unroll
    for (int half = 0; half < 2; ++half) {
      const int c = idx + half * 64;
      const float S_H  = ws[c * 4 + 0];
      const float S_HI = ws[c * 4 + 1];
      const float S_I  = ws[c * 4 + 2];
      const float S_I2 = ws[c * 4 + 3];
      const float rest = S_I - S_HI;
      const float c1 = S_HI / (S_H + EPS_F);
      const float c2 = rest / ((HWf - S_H) + EPS_F);
      f += S_I2 - 2.0f * c1 * S_HI - 2.0f * c2 * rest
         + c1 * c1 * S_H + c2 * c2 * (HWf - S_H);
    }
    fold[j] = f;
  }

  // D = A(16x4 data) x B(4x16 ones) + 0 : D[m][n] = rowsum_m.
  // Per-lane sum of the 8 C/D VGPRs = half the rowsums; lane^16 holds the other half.
  v2f a; a.x = fold[0]; a.y = fold[1];
  v2f b; b.x = 1.0f;    b.y = 1.0f;
  v8f cacc = {};
  v8f d = __builtin_amdgcn_wmma_f32_16x16x4_f32(
      /*neg_a=*/false, a, /*neg_b=*/false, b,
      /*c_mod=*/(short)0, cacc, /*reuse_a=*/false, /*reuse_b=*/false);

  float s = ((d[0] + d[1]) + (d[2] + d[3])) + ((d[4] + d[5]) + (d[6] + d[7]));
  s += __shfl_xor(s, 16);
  if (lane == 0) out[0] = s * (1.0f / (float)(32ull * 4 * 512 * 512));
}

extern "C" void kernel_launch(void* const* d_in, const int* in_sizes, int n_in,
                              void* d_out, int out_size, void* d_ws, size_t ws_size,
                              hipStream_t stream) {
  const float* inp = (const float*)d_in[0];   // 'input'
  const float* img = (const float*)d_in[1];   // 'img_mean'
  float* out = (float*)d_out;
  float* ws  = (float*)d_ws;

  zero_ws_kernel<<<1, NCH * 4, 0, stream>>>(ws);
  partial_sums_kernel<<<NCH * BLOCKS_PER_CH, THREADS, 0, stream>>>(inp, img, ws);
  finalize_kernel<<<1, 32, 0, stream>>>(ws, out);
}